// Decoder_68161130988186
// MI455X (gfx1250) — compile-verified
//
#include <hip/hip_runtime.h>
#include <hip/hip_bf16.h>

typedef float v2f __attribute__((ext_vector_type(2)));
typedef float v4f __attribute__((ext_vector_type(4)));
typedef float v8f __attribute__((ext_vector_type(8)));

#define BATCH 32
#define TCTX  4096
#define EDIM  1024
#define HDIM  1024
#define NTC   8            // t-chunks in value pass
#define TCH   (TCTX / NTC) // 512 t per chunk
#define CHUNK 16           // rows staged per async batch
#define NCHK  (TCH / CHUNK)

// ---------------------------------------------------------------------------
// Kernel 1: QKV projections out = x(32x1024) @ W(1024x1024) via WMMA f32.
// One wave per 16x16 output tile. grid = (N/16, M/16, 3).
// ---------------------------------------------------------------------------
__global__ __launch_bounds__(32) void qkv_wmma_kernel(
    const float* __restrict__ x,
    const float* __restrict__ Wv,
    const float* __restrict__ Wk,
    const float* __restrict__ Wq,
    float* __restrict__ vproj,
    float* __restrict__ kproj,
    float* __restrict__ qproj)
{
    const int nTile = blockIdx.x;   // 0..63
    const int mTile = blockIdx.y;   // 0..1
    const int which = blockIdx.z;   // 0:V 1:K 2:Q

    const float* __restrict__ W   = (which == 0) ? Wv : (which == 1) ? Wk : Wq;
    float*       __restrict__ out = (which == 0) ? vproj : (which == 1) ? kproj : qproj;

    const int lane = threadIdx.x & 31;
    const int half = lane >> 4;
    const int l16  = lane & 15;
    const int m0 = mTile * 16;
    const int n0 = nTile * 16;

    v8f acc = {};
    for (int kk = 0; kk < EDIM; kk += 4) {
        const int ka = kk + 2 * half;
        v2f a, b;
        const float* arow = x + (size_t)(m0 + l16) * EDIM + ka;
        a.x = arow[0];
        a.y = arow[1];
        const float* bcol = W + (size_t)ka * HDIM + n0 + l16;
        b.x = bcol[0];
        b.y = bcol[HDIM];
        acc = __builtin_amdgcn_wmma_f32_16x16x4_f32(
            false, a, false, b, (short)0, acc, false, false);
    }

#pragma unroll
    for (int r = 0; r < 8; ++r) {
        out[(size_t)(m0 + r + 8 * half) * HDIM + n0 + l16] = acc[r];
    }
}

// ---------------------------------------------------------------------------
// Kernel 2: scores[b][t] = dot(key_row(b,t), q(b)), t in [0, TCTX].
// One wave per (b,t); 8 fully-unrolled float4 NT loads per wave.
// ---------------------------------------------------------------------------
__global__ __launch_bounds__(256) void scores_kernel(
    const float* __restrict__ key_cache,
    const float* __restrict__ kproj,
    const float* __restrict__ qproj,
    float* __restrict__ scores)
{
    const int b = blockIdx.x;

    __shared__ float qs[HDIM];
    for (int i = threadIdx.x; i < HDIM; i += blockDim.x)
        qs[i] = qproj[(size_t)b * HDIM + i];
    __syncthreads();

    const int wave = threadIdx.x >> 5;
    const int lane = threadIdx.x & 31;
    const int t = blockIdx.y * 8 + wave;
    if (t > TCTX) return;

    const float* __restrict__ row = (t < TCTX)
        ? key_cache + ((size_t)b * TCTX + t) * HDIM
        : kproj + (size_t)b * HDIM;

    const v4f* __restrict__ row4 = (const v4f*)row;
    const v4f* __restrict__ qs4  = (const v4f*)qs;

    float s = 0.0f;
#pragma unroll
    for (int i = 0; i < HDIM / 4 / 32; ++i) {          // 8 iterations
        const v4f kv = __builtin_nontemporal_load(row4 + lane + 32 * i);
        const v4f qv = qs4[lane + 32 * i];
        s += kv.x * qv.x + kv.y * qv.y + kv.z * qv.z + kv.w * qv.w;
    }

#pragma unroll
    for (int off = 16; off > 0; off >>= 1)
        s += __shfl_xor(s, off, 32);

    if (lane == 0)
        scores[(size_t)b * (TCTX + 1) + t] = s;
}

// ---------------------------------------------------------------------------
// Kernel 3: softmax over scores, IN PLACE -> normalized weights
// p[b][t] = exp(s - max) / (sum * BATCH).  One block per batch.
// ---------------------------------------------------------------------------
__global__ __launch_bounds__(256) void softmax_kernel(float* __restrict__ scores)
{
    const int b   = blockIdx.x;
    const int tid = threadIdx.x;
    float* __restrict__ srow = scores + (size_t)b * (TCTX + 1);

    __shared__ float red[256];

    float m = -1e30f;
    for (int t = tid; t <= TCTX; t += 256)
        m = fmaxf(m, srow[t]);
    red[tid] = m;
    __syncthreads();
#pragma unroll
    for (int s = 128; s > 0; s >>= 1) {
        if (tid < s) red[tid] = fmaxf(red[tid], red[tid + s]);
        __syncthreads();
    }
    const float mx = red[0];
    __syncthreads();

    float psum = 0.0f;
    for (int t = tid; t <= TCTX; t += 256)
        psum += expf(srow[t] - mx);
    red[tid] = psum;
    __syncthreads();
#pragma unroll
    for (int s = 128; s > 0; s >>= 1) {
        if (tid < s) red[tid] += red[tid + s];
        __syncthreads();
    }
    const float inv = 1.0f / (red[0] * (float)BATCH);
    __syncthreads();

    for (int t = tid; t <= TCTX; t += 256)
        srow[t] = expf(srow[t] - mx) * inv;
}

// ---------------------------------------------------------------------------
// Async row stage: one instruction moves one 512 B value row (32 lanes x 16 B)
// from global memory straight into LDS, tracked on ASYNCcnt (no dest VGPRs).
// ---------------------------------------------------------------------------
__device__ __forceinline__ void async_row_load(unsigned lds_off, const float* g)
{
    asm volatile("global_load_async_to_lds_b128 %0, %1, off"
                 :: "v"(lds_off), "v"(g)
                 : "memory");
}

// ---------------------------------------------------------------------------
// Kernel 4: partial weighted value sums with double-buffered async staging.
// grid = (BATCH, HDIM/128, NTC); one wave per block; thread -> 4 h channels.
// Each iteration: issue next 16-row chunk async (16 x 512 B outstanding),
// s_wait_asynccnt 16 guarantees the previous chunk landed (in-order counter),
// then FMA it out of LDS. Up to 16 KB outstanding per wave, zero VGPR cost.
// ---------------------------------------------------------------------------
__global__ __launch_bounds__(32) void av_partial_kernel(
    const float* __restrict__ value_cache,
    const float* __restrict__ p,        // normalized weights (B, TCTX+1)
    float* __restrict__ partial)        // (B, NTC, HDIM)
{
    const int b    = blockIdx.x;
    const int hc   = blockIdx.y;        // 0..7  (128 h each)
    const int tc   = blockIdx.z;        // 0..NTC-1 (TCH t each)
    const int lane = threadIdx.x;       // 0..31
    const int t0   = tc * TCH;

    __shared__ float pw[TCH];
    __shared__ float stage[2][CHUNK * 128];

    for (int i = lane; i < TCH; i += 32)
        pw[i] = p[(size_t)b * (TCTX + 1) + t0 + i];
    __syncthreads();

    // per-lane global base: 16 B segment of row 0 of this block's h-slice
    const float* gbase =
        value_cache + ((size_t)b * TCTX + t0) * HDIM + hc * 128 + lane * 4;

    // per-lane LDS byte offsets of the two stage buffers
    const unsigned lb0 = (unsigned)(size_t)&stage[0][0] + (unsigned)lane * 16u;
    const unsigned lb1 = (unsigned)(size_t)&stage[1][0] + (unsigned)lane * 16u;

    // prologue: stage chunk 0
#pragma unroll
    for (int r = 0; r < CHUNK; ++r)
        async_row_load(lb0 + r * 512u, gbase + (size_t)r * HDIM);

    v4f acc = {};
    for (int c = 0; c < NCHK; ++c) {
        const int buf = c & 1;
        if (c + 1 < NCHK) {
            const unsigned lb = buf ? lb0 : lb1;
#pragma unroll
            for (int r = 0; r < CHUNK; ++r)
                async_row_load(lb + r * 512u,
                               gbase + (size_t)((c + 1) * CHUNK + r) * HDIM);
            asm volatile("s_wait_asynccnt 16" ::: "memory");
        } else {
            asm volatile("s_wait_asynccnt 0" ::: "memory");
        }
#pragma unroll
        for (int r = 0; r < CHUNK; ++r) {
            const float wgt = pw[c * CHUNK + r];
            const v4f v = *(const v4f*)&stage[buf][r * 128 + lane * 4];
            acc += wgt * v;
        }
    }

    *(v4f*)(partial + ((size_t)(b * NTC + tc)) * HDIM + hc * 128 + lane * 4) = acc;
}

// ---------------------------------------------------------------------------
// Kernel 5: reduce partials + appended-v term + residual add.
// ---------------------------------------------------------------------------
__global__ __launch_bounds__(256) void av_final_kernel(
    const float* __restrict__ partial,
    const float* __restrict__ vproj,
    const float* __restrict__ p,
    const float* __restrict__ x,
    float* __restrict__ out)
{
    const int idx = blockIdx.x * 256 + threadIdx.x;   // 0 .. B*H-1
    const int b = idx >> 10;
    const int h = idx & (HDIM - 1);

    float acc = 0.0f;
#pragma unroll
    for (int c = 0; c < NTC; ++c)
        acc += partial[((size_t)(b * NTC + c)) * HDIM + h];

    acc += p[(size_t)b * (TCTX + 1) + TCTX] * vproj[(size_t)b * HDIM + h];
    out[idx] = acc + x[idx];
}

// ---------------------------------------------------------------------------
extern "C" void kernel_launch(void* const* d_in, const int* in_sizes, int n_in,
                              void* d_out, int out_size, void* d_ws, size_t ws_size,
                              hipStream_t stream) {
    const float* x    = (const float*)d_in[0];   // (32, 1024)
    const float* keyc = (const float*)d_in[1];   // (32, 4096, 1024)
    const float* valc = (const float*)d_in[2];   // (32, 4096, 1024)
    const float* Wv   = (const float*)d_in[3];   // (1024, 1024)
    const float* Wk   = (const float*)d_in[4];
    const float* Wq   = (const float*)d_in[5];
    float* out = (float*)d_out;                  // (32, 1024)

    float* ws      = (float*)d_ws;
    float* qproj   = ws;                              // 32*1024
    float* kproj   = ws + BATCH * HDIM;               // 32*1024
    float* vproj   = ws + 2 * BATCH * HDIM;           // 32*1024
    float* scores  = ws + 3 * BATCH * HDIM;           // 32*4097 (becomes p)
    float* partial = scores + BATCH * (TCTX + 1);     // 32*NTC*1024

    // 1) QKV projections (v_wmma_f32_16x16x4_f32)
    qkv_wmma_kernel<<<dim3(HDIM / 16, BATCH / 16, 3), 32, 0, stream>>>(
        x, Wv, Wk, Wq, vproj, kproj, qproj);

    // 2) attention scores over key cache + appended k row
    scores_kernel<<<dim3(BATCH, (TCTX + 1 + 7) / 8), 256, 0, stream>>>(
        keyc, kproj, qproj, scores);

    // 3) softmax -> normalized weights in place (includes 1/32)
    softmax_kernel<<<dim3(BATCH), 256, 0, stream>>>(scores);

    // 4) partial weighted value sums (async global->LDS staging)
    av_partial_kernel<<<dim3(BATCH, HDIM / 128, NTC), 32, 0, stream>>>(
        valc, scores, partial);

    // 5) deterministic reduction + residual
    av_final_kernel<<<dim3((BATCH * HDIM) / 256), 256, 0, stream>>>(
        partial, vproj, scores, x, out);
}